// LigerMHC_44032004719154
// MI455X (gfx1250) — compile-verified
//
#include <hip/hip_runtime.h>
#include <math.h>

typedef __attribute__((ext_vector_type(2))) float v2f;
typedef __attribute__((ext_vector_type(4))) float v4f;
typedef __attribute__((ext_vector_type(8))) float v8f;

#define HC 4
#define CDIM 256
#define KDIM 1024          // HC*C
#define MCO 24             // HC*HC + 2*HC
#define TMAXI 20
#define TOKW 16            // tokens per wave (one 16-row WMMA tile)

// ---------------------------------------------------------------------------
// Prep: repack phi (1024x24, zero-pad to 32 cols) and W^T (B[k][n] = W[n][k])
// into per-lane WMMA B-fragment order for V_WMMA_F32_16X16X4_F32:
//   entry[(ntile, kstep, lane)] = { B[k+koff][n], B[k+koff+1][n] }
//   n = ntile*16 + (lane&15), koff = (lane>>4)*2, k = kstep*4
// phiP: 2*256*32 v2f (128 KB)   WP: 16*64*32 v2f (256 KB)  -> 384 KB of d_ws
// ---------------------------------------------------------------------------
__global__ void mhc_repack(const float* __restrict__ phi, const float* __restrict__ W,
                           v2f* __restrict__ phiP, v2f* __restrict__ WP) {
  int idx = blockIdx.x * blockDim.x + threadIdx.x;
  const int NPHI = 2 * 256 * 32;
  const int NW   = 16 * 64 * 32;
  if (idx < NPHI) {
    int nt = idx / (256 * 32);
    int rem = idx % (256 * 32);
    int ks = rem / 32;
    int lane = rem % 32;
    int n = nt * 16 + (lane & 15);
    int k = ks * 4 + ((lane >> 4) * 2);
    v2f v;
    v.x = (n < MCO) ? phi[k * MCO + n] : 0.0f;
    v.y = (n < MCO) ? phi[(k + 1) * MCO + n] : 0.0f;
    phiP[idx] = v;
  } else if (idx < NPHI + NW) {
    int j = idx - NPHI;
    int nt = j / (64 * 32);
    int rem = j % (64 * 32);
    int ks = rem / 32;
    int lane = rem % 32;
    int n = nt * 16 + (lane & 15);
    int k = ks * 4 + ((lane >> 4) * 2);
    v2f v;
    v.x = W[n * CDIM + k];       // B[k][n]   = W[n][k]
    v.y = W[n * CDIM + k + 1];   // B[k+1][n] = W[n][k+1]
    WP[j] = v;
  }
}

// ---------------------------------------------------------------------------
// Fused kernel: one wave (32 lanes) handles 16 tokens.
//   1) stage x tile (16x1024 f32) to LDS via global_load_async_to_lds_b128
//   2) logits = V @ phi via v_wmma_f32_16x16x4_f32 (2 n-tiles x 256 k-steps)
//   3) lanes 0-15: RMS + bias + sigmoid + 20x Sinkhorn (per token)
//   4) x_in = sum_i h_pre[i] * x[i,:]  (LDS)
//   5) f = x_in @ W^T via WMMA (16 n-tiles x 64 k-steps)
//   6) out = P @ x + h_post * f   (v4f, coalesced non-temporal stores)
// ---------------------------------------------------------------------------
__global__ __launch_bounds__(32) void mhc_fused(
    const float* __restrict__ X, const v2f* __restrict__ phiP,
    const v2f* __restrict__ WP, const float* __restrict__ bv,
    const float* __restrict__ apre_p, const float* __restrict__ apost_p,
    const float* __restrict__ ares_p, float* __restrict__ OUT, int nTok) {
  __shared__ __align__(16) float lds_x[TOKW * KDIM];     // 64 KB
  __shared__ __align__(16) float lds_xin[TOKW * CDIM];   // 16 KB
  __shared__ __align__(16) float lds_f[TOKW * CDIM];     // 16 KB
  __shared__ float lds_coef[TOKW * MCO];                 // 1.5 KB

  const int lane = threadIdx.x & 31;
  const long t0 = (long)blockIdx.x * TOKW;
  if (t0 >= nTok) return;

  // ---- 1) stage x tile: async global -> LDS (ASYNCcnt path, no VGPR bounce) ----
  {
    const v4f* gsrc = (const v4f*)(X + (size_t)t0 * KDIM) + lane;
    // flat shared-pointer low 32 bits == wave-relative LDS byte offset
    unsigned ldsa = (unsigned)(size_t)(void*)(((v4f*)lds_x) + lane);
    for (int it = 0; it < (TOKW * KDIM) / (4 * 32); ++it) {  // 128 iters
      unsigned long long ga = (unsigned long long)(size_t)(gsrc + it * 32);
      asm volatile("global_load_async_to_lds_b128 %0, %1, off"
                   :: "v"(ldsa + it * 512u), "v"(ga)
                   : "memory");
    }
    asm volatile("s_wait_asynccnt 0x0" ::: "memory");
  }
  __syncthreads();

  const int row  = lane & 15;        // M (token) for A-frags, N for B-frags
  const int koff = (lane >> 4) * 2;  // K sub-offset for A/B frags
  const int mbase = (lane >> 4) * 8; // M base for C/D frags

  // ---- 2) logits WMMA: L(16x24) = V(16x1024) @ phi(1024x24) ----
  v8f acc0 = {};  // cols 0..15
  v8f acc1 = {};  // cols 16..23 (24..31 padded zero)
  {
    const float* xr = lds_x + row * KDIM;
    for (int kk = 0; kk < KDIM; kk += 4) {
      v2f a = *(const v2f*)(xr + kk + koff);
      v2f b0 = phiP[(kk >> 2) * 32 + lane];
      v2f b1 = phiP[(256 + (kk >> 2)) * 32 + lane];
      acc0 = __builtin_amdgcn_wmma_f32_16x16x4_f32(false, a, false, b0,
                                                   (short)0, acc0, false, false);
      acc1 = __builtin_amdgcn_wmma_f32_16x16x4_f32(false, a, false, b1,
                                                   (short)0, acc1, false, false);
    }
  }
  // spill logit frags to LDS: C/D layout, VGPR j -> M = mbase+j, N = lane&15
#pragma unroll
  for (int j = 0; j < 8; ++j) {
    int m = mbase + j;
    lds_coef[m * MCO + row] = acc0[j];
    if (row < 8) lds_coef[m * MCO + 16 + row] = acc1[j];
  }
  __syncthreads();

  // ---- 3) per-token scalar stage: lanes 0..15, one token each ----
  const float a_pre = apre_p[0], a_post = apost_p[0], a_res = ares_p[0];
  if (lane < 16) {
    float l[MCO];
#pragma unroll
    for (int i = 0; i < MCO; ++i) l[i] = lds_coef[lane * MCO + i];
    float ss = 0.0f;
#pragma unroll
    for (int i = 0; i < MCO; ++i) ss += l[i] * l[i];
    float rinv = 1.0f / sqrtf(ss * (1.0f / MCO) + 1e-6f);
    float y[MCO];
#pragma unroll
    for (int i = 0; i < MCO; ++i) y[i] = l[i] * rinv + bv[i];
    float hpre[4], hpost[4], P[16];
#pragma unroll
    for (int i = 0; i < 4; ++i) hpre[i] = 1.0f / (1.0f + __expf(-a_pre * y[i]));
#pragma unroll
    for (int i = 0; i < 4; ++i) hpost[i] = 2.0f / (1.0f + __expf(-a_post * y[4 + i]));
#pragma unroll
    for (int i = 0; i < 16; ++i) P[i] = __expf(a_res * y[8 + i]);
    for (int it = 0; it < TMAXI; ++it) {
#pragma unroll
      for (int r = 0; r < 4; ++r) {
        float s = P[4 * r] + P[4 * r + 1] + P[4 * r + 2] + P[4 * r + 3] + 1e-6f;
        float si = 1.0f / s;
        P[4 * r] *= si; P[4 * r + 1] *= si; P[4 * r + 2] *= si; P[4 * r + 3] *= si;
      }
#pragma unroll
      for (int c = 0; c < 4; ++c) {
        float s = P[c] + P[4 + c] + P[8 + c] + P[12 + c] + 1e-6f;
        float si = 1.0f / s;
        P[c] *= si; P[4 + c] *= si; P[8 + c] *= si; P[12 + c] *= si;
      }
    }
#pragma unroll
    for (int i = 0; i < 4; ++i)  lds_coef[lane * MCO + i] = hpre[i];
#pragma unroll
    for (int i = 0; i < 4; ++i)  lds_coef[lane * MCO + 4 + i] = hpost[i];
#pragma unroll
    for (int i = 0; i < 16; ++i) lds_coef[lane * MCO + 8 + i] = P[i];
  }
  __syncthreads();

  // ---- 4) x_in[m][c] = sum_i h_pre[m][i] * x[m][i*256+c] ----
#pragma unroll 2
  for (int e = lane; e < TOKW * CDIM; e += 32) {
    int m = e >> 8;
    int c = e & 255;
    const float* xm = lds_x + m * KDIM;
    float h0 = lds_coef[m * MCO + 0], h1 = lds_coef[m * MCO + 1];
    float h2 = lds_coef[m * MCO + 2], h3 = lds_coef[m * MCO + 3];
    lds_xin[e] = h0 * xm[c] + h1 * xm[256 + c] + h2 * xm[512 + c] + h3 * xm[768 + c];
  }
  __syncthreads();

  // ---- 5) f(16x256) = x_in(16x256) @ W^T via WMMA ----
  {
    const float* xinr = lds_xin + row * CDIM;
    for (int nt = 0; nt < 16; ++nt) {
      v8f acc = {};
      for (int ks = 0; ks < 64; ++ks) {
        v2f a = *(const v2f*)(xinr + ks * 4 + koff);
        v2f b = WP[(nt * 64 + ks) * 32 + lane];
        acc = __builtin_amdgcn_wmma_f32_16x16x4_f32(false, a, false, b,
                                                    (short)0, acc, false, false);
      }
#pragma unroll
      for (int j = 0; j < 8; ++j)
        lds_f[(mbase + j) * CDIM + nt * 16 + row] = acc[j];
    }
  }
  __syncthreads();

  // ---- 6) out[m][o][c] = sum_i P[m][o][i]*x[m][i][c] + h_post[m][o]*f[m][c] ----
  // Output is written once and never re-read: use non-temporal stores so L2
  // capacity stays reserved for the reused phi/W fragments and x tiles.
  v4f* out4 = (v4f*)(OUT + (size_t)t0 * KDIM);
  for (int e = lane; e < TOKW * CDIM; e += 32) {  // 4096 v4f per tile
    int m  = e >> 8;          // token in tile
    int w  = e & 255;         // v4f index within token (256 per token)
    int o  = w >> 6;          // output head
    int c4 = w & 63;          // v4f column
    const v4f* xm = (const v4f*)(lds_x + m * KDIM);
    const v4f* fm = (const v4f*)(lds_f + m * CDIM);
    float p0 = lds_coef[m * MCO + 8 + o * 4 + 0];
    float p1 = lds_coef[m * MCO + 8 + o * 4 + 1];
    float p2 = lds_coef[m * MCO + 8 + o * 4 + 2];
    float p3 = lds_coef[m * MCO + 8 + o * 4 + 3];
    float hp = lds_coef[m * MCO + 4 + o];
    v4f x0 = xm[c4], x1 = xm[64 + c4], x2 = xm[128 + c4], x3 = xm[192 + c4];
    v4f f = fm[c4];
    v4f r = p0 * x0 + p1 * x1 + p2 * x2 + p3 * x3 + hp * f;
    __builtin_nontemporal_store(r, &out4[e]);
  }
}

extern "C" void kernel_launch(void* const* d_in, const int* in_sizes, int n_in,
                              void* d_out, int out_size, void* d_ws, size_t ws_size,
                              hipStream_t stream) {
  const float* X     = (const float*)d_in[0];
  const float* phi   = (const float*)d_in[1];
  const float* bv    = (const float*)d_in[2];
  const float* apre  = (const float*)d_in[3];
  const float* apost = (const float*)d_in[4];
  const float* ares  = (const float*)d_in[5];
  const float* W     = (const float*)d_in[6];
  float* OUT = (float*)d_out;

  v2f* phiP = (v2f*)d_ws;            // 128 KB
  v2f* WP   = phiP + 2 * 256 * 32;   // 256 KB

  int nTok = in_sizes[0] / KDIM;     // 32768

  int total = 2 * 256 * 32 + 16 * 64 * 32;
  mhc_repack<<<(total + 255) / 256, 256, 0, stream>>>(phi, W, phiP, WP);

  int nBlocks = (nTok + TOKW - 1) / TOKW;  // 2048
  mhc_fused<<<nBlocks, 32, 0, stream>>>(X, phiP, WP, bv, apre, apost, ares, OUT,
                                        nTok);
}